// KAST_90787018703342
// MI455X (gfx1250) — compile-verified
//
#include <hip/hip_runtime.h>
#include <math.h>

typedef float v2f __attribute__((ext_vector_type(2)));
typedef float v8f __attribute__((ext_vector_type(8)));

#define COEF_MEM 0.2f

__device__ __forceinline__ float lane_reduce_max16(float x) {
#pragma unroll
  for (int m = 1; m <= 8; m <<= 1) x = fmaxf(x, __shfl_xor(x, m, 32));
  return x;
}
__device__ __forceinline__ float lane_reduce_sum16(float x) {
#pragma unroll
  for (int m = 1; m <= 8; m <<= 1) x += __shfl_xor(x, m, 32);
  return x;
}

// One flash-attention pass over n_tiles tiles of 16 reference rows.
// qlds: this wave's 16x256 fp32 Q tile in LDS.
// refg: [n_tiles*16][256] keys, vg: [n_tiles*16][3] values.
// res[r][d]: softmax(Q Ref^T) @ V for rows M = r and M = r+8 (per lane half).
__device__ __forceinline__ void flash_pass(
    const float* __restrict__ qlds,
    const float* __restrict__ refg,
    const float* __restrict__ vg,
    int n_tiles, int lane,
    float res[8][3])
{
  const int rlo   = lane & 15;   // N (ref row in tile) / M (q row) index
  const int cbase = (lane >> 4) * 2;  // channel sub-offset for A/B fragments

  float runmax[8], runsum[8], acc[8][3];
#pragma unroll
  for (int r = 0; r < 8; ++r) {
    runmax[r] = -3.0e38f;
    runsum[r] = 0.0f;
    acc[r][0] = acc[r][1] = acc[r][2] = 0.0f;
  }

  for (int t = 0; t < n_tiles; ++t) {
    const int n0 = t * 16;
    const float* refrow = refg + (size_t)(n0 + rlo) * 256;
    const float* vrow   = vg   + (size_t)(n0 + rlo) * 3;
    const float vv0 = vrow[0], vv1 = vrow[1], vv2 = vrow[2];

    // S tile = Q(16x256) * Ref^T(256x16), accumulated fp32 via WMMA K=4 steps
    v8f cacc = {0.f, 0.f, 0.f, 0.f, 0.f, 0.f, 0.f, 0.f};
#pragma unroll 8
    for (int j = 0; j < 64; ++j) {
      v2f a = *(const v2f*)(qlds + rlo * 256 + j * 4 + cbase);   // A from LDS
      v2f b = *(const v2f*)(refrow + j * 4 + cbase);             // B from global
      cacc = __builtin_amdgcn_wmma_f32_16x16x4_f32(
          /*neg_a=*/false, a, /*neg_b=*/false, b,
          /*c_mod=*/(short)0, cacc, /*reuse_a=*/false, /*reuse_b=*/false);
    }

    // Online softmax + P*V, fused in the lane reduction (cv = 3)
#pragma unroll
    for (int r = 0; r < 8; ++r) {
      float s    = cacc[r];
      float tmax = lane_reduce_max16(s);
      float nmax = fmaxf(runmax[r], tmax);
      float corr = __expf(runmax[r] - nmax);
      float p    = __expf(s - nmax);
      float psum = lane_reduce_sum16(p);
      float pv0  = lane_reduce_sum16(p * vv0);
      float pv1  = lane_reduce_sum16(p * vv1);
      float pv2  = lane_reduce_sum16(p * vv2);
      runmax[r] = nmax;
      runsum[r] = runsum[r] * corr + psum;
      acc[r][0] = acc[r][0] * corr + pv0;
      acc[r][1] = acc[r][1] * corr + pv1;
      acc[r][2] = acc[r][2] * corr + pv2;
    }
  }

#pragma unroll
  for (int r = 0; r < 8; ++r) {
    float inv = 1.0f / runsum[r];
    res[r][0] = acc[r][0] * inv;
    res[r][1] = acc[r][1] * inv;
    res[r][2] = acc[r][2] * inv;
  }
}

// grid: 56 (b,t) pairs * 16 q-blocks = 896 WGs; 128 threads = 4 waves;
// wave w handles q rows [qblk*64 + w*16, +16)
__global__ __launch_bounds__(128) void KAST_attn_kernel(
    const float* __restrict__ k,
    const float* __restrict__ v,
    const float* __restrict__ mk,
    const float* __restrict__ mv,
    float* __restrict__ out)
{
  __shared__ float qlds[4][16 * 256];  // 64 KB

  const int wave = threadIdx.x >> 5;
  const int lane = threadIdx.x & 31;
  const int bt   = blockIdx.x >> 4;   // 0..55
  const int qblk = blockIdx.x & 15;
  const int b  = bt / 7;
  const int tq = bt % 7 + 1;          // query time index
  const int tr = tq - 1;              // reference time index

  const int qbase = qblk * 64 + wave * 16;
  const float* qg = k + (((size_t)(b * 8 + tq)) * 1024 + qbase) * 256;

  // Stage this wave's Q tile into LDS (16 rows x 256 f32 = 1024 float4)
  float* qs = qlds[wave];
  for (int i = lane; i < 16 * 64; i += 32) {
    ((float4*)qs)[i] = ((const float4*)qg)[i];
  }
  __syncthreads();

  const float* refg = k  + ((size_t)(b * 8 + tr)) * 1024 * 256;
  const float* vg   = v  + ((size_t)(b * 8 + tr)) * 1024 * 3;
  const float* mkg  = mk + ((size_t)(b * 8 + tr)) * 512 * 256;
  const float* mvg  = mv + ((size_t)(b * 8 + tr)) * 512 * 3;

  float res1[8][3], res2[8][3];
  flash_pass(qs, refg, vg, 64, lane, res1);   // spatial keys: 1024 rows
  flash_pass(qs, mkg, mvg, 32, lane, res2);   // memory keys:   512 rows

  const int rlo = lane & 15;
  const int hi  = lane >> 4;
  float* outg = out + (((size_t)(b * 7 + tr)) * 1024 + qbase) * 3;
#pragma unroll
  for (int r = 0; r < 8; ++r) {
    if (rlo == r) {  // one lane per row per half-wave holds the final value
      const int row = r + 8 * hi;
      outg[row * 3 + 0] = (1.0f - COEF_MEM) * res1[r][0] + COEF_MEM * res2[r][0];
      outg[row * 3 + 1] = (1.0f - COEF_MEM) * res1[r][1] + COEF_MEM * res2[r][1];
      outg[row * 3 + 2] = (1.0f - COEF_MEM) * res1[r][2] + COEF_MEM * res2[r][2];
    }
  }
}

// ground_truth = v[:, 1:]  (pure copy with time shift)
__global__ void KAST_gt_copy(const float* __restrict__ v, float* __restrict__ gt)
{
  const int N = 8 * 7 * 1024 * 3;
  int i = blockIdx.x * 256 + threadIdx.x;
  if (i >= N) return;
  int d   = i % 3;
  int row = (i / 3) % 1024;
  int bt  = i / (3 * 1024);          // 0..55
  int b = bt / 7, t = bt % 7 + 1;
  gt[i] = v[(((size_t)(b * 8 + t)) * 1024 + row) * 3 + d];
}

extern "C" void kernel_launch(void* const* d_in, const int* in_sizes, int n_in,
                              void* d_out, int out_size, void* d_ws, size_t ws_size,
                              hipStream_t stream) {
  const float* k  = (const float*)d_in[0];
  const float* v  = (const float*)d_in[1];
  const float* mk = (const float*)d_in[2];
  const float* mv = (const float*)d_in[3];
  float* out = (float*)d_out;

  KAST_attn_kernel<<<dim3(896), dim3(128), 0, stream>>>(k, v, mk, mv, out);

  const int N = 8 * 7 * 1024 * 3;
  KAST_gt_copy<<<dim3((N + 255) / 256), dim3(256), 0, stream>>>(v, out + N);
}